// RotarySelfAttention_23940147708412
// MI455X (gfx1250) — compile-verified
//
#include <hip/hip_runtime.h>
#include <hip/hip_bf16.h>
#include <math.h>

// ---------------------------------------------------------------------------
// RotarySelfAttention (rope numerator / raw denominator linear-style attn),
// fully fused on gfx1250 using V_WMMA_F32_16X16X4_F32.
// B=2, N=2048, D=1024, H=16, hd=64.
//
// K1: qkv GEMM + bias + elu+1, scatter to (B,H,N,hd) q/k/v scratch.
// K2: flash-style  num=rope(q)@rope(k)^T, den=q@k^T, W=num/den, O+=W@V,
//     with rope applied on the fly (register-level for q-fragments,
//     during LDS staging for k-tiles).
// K3: output projection GEMM + bias.
// ---------------------------------------------------------------------------

typedef __attribute__((ext_vector_type(2))) float v2f;
typedef __attribute__((ext_vector_type(8))) float v8f;

__device__ __forceinline__ v8f wmma4(v2f a, v2f b, v8f c) {
  // D(16x16,f32) = A(16x4,f32) x B(4x16,f32) + C
  return __builtin_amdgcn_wmma_f32_16x16x4_f32(
      false, a, false, b, (short)0, c, false, false);
}

#define NB   2
#define NSEQ 2048
#define DIM  1024
#define NH   16
#define HD   64
#define D3   3072
#define LN10000_OVER_32 0.28782313662425573f  // ln(10000)/32

// ---------------------------------------------------------------------------
// Kernel 1: qkv = x @ w_qkv + b_qkv, fused split/heads/elu+1.
// One wave computes a 16(row) x 64(col) strip (4 C tiles, shared A frag).
// Grid: 256 rowTiles * 48 colGroups waves, 4 waves / block -> 3072 blocks.
// ---------------------------------------------------------------------------
__global__ __launch_bounds__(128) void qkv_kernel(
    const float* __restrict__ x, const float* __restrict__ wqkv,
    const float* __restrict__ bqkv, float* __restrict__ qb,
    float* __restrict__ kb, float* __restrict__ vb) {
  const int wave = threadIdx.x >> 5;
  const int lane = threadIdx.x & 31;
  const int rlo  = lane & 15;
  const int kh   = lane >> 4;                 // 0 or 1 (K half)
  const int wg   = blockIdx.x * 4 + wave;
  const int rowTile  = wg / 48;
  const int colGroup = wg % 48;
  const int row0 = rowTile * 16;
  const int col0 = colGroup * 64;

  v8f acc[4] = {};
  const float* arow = x + (size_t)(row0 + rlo) * DIM;

  #pragma unroll 2
  for (int k0 = 0; k0 < DIM; k0 += 4) {
    const int kk = k0 + kh * 2;
    v2f a;
    a.x = arow[kk];
    a.y = arow[kk + 1];
    const float* bp = wqkv + (size_t)kk * D3 + col0 + rlo;
    #pragma unroll
    for (int t = 0; t < 4; ++t) {
      v2f b;
      b.x = bp[t * 16];
      b.y = bp[D3 + t * 16];
      acc[t] = wmma4(a, b, acc[t]);
    }
  }

  #pragma unroll
  for (int t = 0; t < 4; ++t) {
    const int col  = col0 + t * 16 + rlo;
    const float bias = bqkv[col];
    const int part = col >> 10;               // 0=q 1=k 2=v (wave-uniform)
    const int d    = col & 1023;
    const int h    = d >> 6;
    const int dh   = d & 63;
    float* dst = (part == 0) ? qb : (part == 1) ? kb : vb;
    #pragma unroll
    for (int i = 0; i < 8; ++i) {
      const int r  = row0 + i + 8 * kh;
      const int bb = r >> 11;                 // N = 2048
      const int n  = r & 2047;
      const size_t hidx = ((size_t)(bb * NH + h) * NSEQ + n) * HD + dh;
      const float val = acc[t][i] + bias;
      // elu(x)+1 : x>0 -> x+1 ; x<=0 -> exp(x)   (v passes through raw)
      dst[hidx] = (part == 2) ? val
                              : (val > 0.0f ? val + 1.0f : expf(val));
    }
  }
}

// ---------------------------------------------------------------------------
// Kernel 2: flash-style fused attention with on-the-fly rope.
// Block = 8 waves, all on one (b,h); each wave owns 16 queries (16x64 acc).
// k/v 16x64 key tiles streamed through LDS (coalesced float4); the rotated
// rk tile is produced in registers during staging (pairs live in the float4).
// Grid: 32 (b,h) * 16 query chunks = 512 blocks.
// ---------------------------------------------------------------------------
__global__ __launch_bounds__(256) void attn_kernel(
    const float* __restrict__ qb, const float* __restrict__ kb,
    const float* __restrict__ vb, float* __restrict__ oh) {
  __shared__ float ks[16 * 64];
  __shared__ float rks[16 * 64];
  __shared__ float vs[16 * 64];
  __shared__ float wls[8 * 16 * 17];          // per-wave W staging, pad 17

  const int wave = threadIdx.x >> 5;
  const int lane = threadIdx.x & 31;
  const int rlo  = lane & 15;
  const int kh   = lane >> 4;
  const int bh     = blockIdx.x >> 4;
  const int qChunk = blockIdx.x & 15;
  const int n0     = qChunk * 128 + wave * 16;
  const size_t base = (size_t)bh * NSEQ * HD;

  // Preload 16x64 q A-fragments; build rotated-q fragments in registers:
  // each fragment holds the (even,odd) rope pair (d = kk*4 + kh*2 is even).
  v2f aq[16], arq[16];
  {
    const float nq = (float)(n0 + rlo);
    #pragma unroll
    for (int kk = 0; kk < 16; ++kk) {
      const int d = kk * 4 + kh * 2;
      const size_t idx = base + (size_t)(n0 + rlo) * HD + d;
      v2f a;
      a.x = qb[idx];
      a.y = qb[idx + 1];
      aq[kk] = a;
      const float theta = expf(-LN10000_OVER_32 * (float)(d >> 1));
      float sn, cs;
      sincosf(nq * theta, &sn, &cs);
      arq[kk].x = a.x * cs - a.y * sn;
      arq[kk].y = a.x * sn + a.y * cs;
    }
  }

  // Cooperative-staging geometry: thread tid owns key row jl, dims d0..d0+3
  // (two complete rope pairs). Thetas are loop-invariant.
  const int tid = threadIdx.x;
  const int jl  = tid >> 4;                   // 0..15 key row in tile
  const int d0  = (tid & 15) * 4;             // 0..60
  const float theta0 = expf(-LN10000_OVER_32 * (float)(d0 >> 1));
  const float theta1 = expf(-LN10000_OVER_32 * (float)((d0 >> 1) + 1));

  v8f o0 = {}, o1 = {}, o2 = {}, o3 = {};
  float* wl = wls + wave * (16 * 17);

  for (int jt = 0; jt < NSEQ / 16; ++jt) {
    {  // stage k (raw + rotated) and v tiles: 1 float4 per thread each
      const size_t g = base + (size_t)jt * 16 * HD;
      const float4 kv = ((const float4*)(kb + g))[tid];
      const float4 vv = ((const float4*)(vb + g))[tid];
      const float nk = (float)(jt * 16 + jl);
      float s0, c0, s1, c1;
      sincosf(nk * theta0, &s0, &c0);
      sincosf(nk * theta1, &s1, &c1);
      float4 rv;
      rv.x = kv.x * c0 - kv.y * s0;
      rv.y = kv.x * s0 + kv.y * c0;
      rv.z = kv.z * c1 - kv.w * s1;
      rv.w = kv.z * s1 + kv.w * c1;
      ((float4*)ks)[tid]  = kv;
      ((float4*)rks)[tid] = rv;
      ((float4*)vs)[tid]  = vv;
      if (jt + 1 < NSEQ / 16) {               // pull next tile toward L2/L0
        const size_t gn = g + 16 * HD;
        __builtin_prefetch(kb + gn + tid * 4, 0, 3);
        __builtin_prefetch(vb + gn + tid * 4, 0, 3);
      }
    }
    __syncthreads();

    // num / den 16x16 tiles over K = 64
    v8f num = {}, den = {};
    #pragma unroll
    for (int kk = 0; kk < 16; ++kk) {
      const int dd = kk * 4 + kh * 2;
      v2f bk, brk;
      bk.x  = ks[rlo * 64 + dd];   bk.y  = ks[rlo * 64 + dd + 1];
      brk.x = rks[rlo * 64 + dd];  brk.y = rks[rlo * 64 + dd + 1];
      num = wmma4(arq[kk], brk, num);
      den = wmma4(aq[kk],  bk,  den);
    }

    // W = num / den, staged to LDS (C layout write -> A layout read)
    #pragma unroll
    for (int i = 0; i < 8; ++i) {
      wl[(i + 8 * kh) * 17 + rlo] = num[i] / den[i];
    }
    // O(16x64) += W(16x16) @ V(16x64); wave-private LDS ops are in-order.
    #pragma unroll
    for (int kk2 = 0; kk2 < 4; ++kk2) {
      const int jj = kk2 * 4 + kh * 2;
      v2f a;
      a.x = wl[rlo * 17 + jj];
      a.y = wl[rlo * 17 + jj + 1];
      v2f b;
      b.x = vs[jj * 64 + rlo];        b.y = vs[(jj + 1) * 64 + rlo];
      o0 = wmma4(a, b, o0);
      b.x = vs[jj * 64 + 16 + rlo];   b.y = vs[(jj + 1) * 64 + 16 + rlo];
      o1 = wmma4(a, b, o1);
      b.x = vs[jj * 64 + 32 + rlo];   b.y = vs[(jj + 1) * 64 + 32 + rlo];
      o2 = wmma4(a, b, o2);
      b.x = vs[jj * 64 + 48 + rlo];   b.y = vs[(jj + 1) * 64 + 48 + rlo];
      o3 = wmma4(a, b, o3);
    }
    __syncthreads();
  }

  #pragma unroll
  for (int i = 0; i < 8; ++i) {
    const size_t r = base + (size_t)(n0 + i + 8 * kh) * HD;
    oh[r + rlo]      = o0[i];
    oh[r + 16 + rlo] = o1[i];
    oh[r + 32 + rlo] = o2[i];
    oh[r + 48 + rlo] = o3[i];
  }
}

// ---------------------------------------------------------------------------
// Kernel 3: out = reshape(out_heads) @ w_out + b_out  (4096 x 1024 x 1024).
// A gathered from (B,H,N,hd) head layout. 16x64 strip per wave.
// Grid: 256 rowTiles * 16 colGroups / 4 waves = 1024 blocks.
// ---------------------------------------------------------------------------
__global__ __launch_bounds__(128) void proj_kernel(
    const float* __restrict__ oh, const float* __restrict__ wout,
    const float* __restrict__ bout, float* __restrict__ out) {
  const int wave = threadIdx.x >> 5;
  const int lane = threadIdx.x & 31;
  const int rlo  = lane & 15;
  const int kh   = lane >> 4;
  const int wg   = blockIdx.x * 4 + wave;
  const int rowTile  = wg >> 4;
  const int colGroup = wg & 15;
  const int row0 = rowTile * 16;
  const int col0 = colGroup * 64;
  const int r  = row0 + rlo;
  const int bb = r >> 11;
  const int n  = r & 2047;

  v8f acc[4] = {};
  #pragma unroll 2
  for (int k0 = 0; k0 < DIM; k0 += 4) {
    const int c  = k0 + kh * 2;
    const int h  = c >> 6;
    const int dh = c & 63;                    // c even -> pair stays in head
    const size_t aidx = ((size_t)(bb * NH + h) * NSEQ + n) * HD + dh;
    v2f a;
    a.x = oh[aidx];
    a.y = oh[aidx + 1];
    const float* bp = wout + (size_t)c * DIM + col0 + rlo;
    #pragma unroll
    for (int t = 0; t < 4; ++t) {
      v2f b;
      b.x = bp[t * 16];
      b.y = bp[DIM + t * 16];
      acc[t] = wmma4(a, b, acc[t]);
    }
  }

  #pragma unroll
  for (int t = 0; t < 4; ++t) {
    const int col = col0 + t * 16 + rlo;
    const float bias = bout[col];
    #pragma unroll
    for (int i = 0; i < 8; ++i) {
      const int rr = row0 + i + 8 * kh;
      out[(size_t)rr * DIM + col] = acc[t][i] + bias;
    }
  }
}

// ---------------------------------------------------------------------------
extern "C" void kernel_launch(void* const* d_in, const int* in_sizes, int n_in,
                              void* d_out, int out_size, void* d_ws,
                              size_t ws_size, hipStream_t stream) {
  const float* x     = (const float*)d_in[0];
  const float* wqkv  = (const float*)d_in[1];
  const float* bqkv  = (const float*)d_in[2];
  const float* wout  = (const float*)d_in[3];
  const float* bout  = (const float*)d_in[4];
  float* out = (float*)d_out;

  // Workspace: 4 buffers of B*H*N*hd f32 each (q, k, v, out_heads) = 67 MB
  const size_t SZ = (size_t)NB * NH * NSEQ * HD;  // 4,194,304 floats
  float* ws = (float*)d_ws;
  float* qb = ws + 0 * SZ;
  float* kb = ws + 1 * SZ;
  float* vb = ws + 2 * SZ;
  float* oh = ws + 3 * SZ;

  // 256 rowTiles * 48 colGroups waves / 4 waves per block
  qkv_kernel<<<3072, 128, 0, stream>>>(x, wqkv, bqkv, qb, kb, vb);
  // 32 (b,h) * 16 query chunks
  attn_kernel<<<512, 256, 0, stream>>>(qb, kb, vb, oh);
  // 256 rowTiles * 16 colGroups waves / 4 waves per block
  proj_kernel<<<1024, 128, 0, stream>>>(oh, wout, bout, out);
}